// MPNN_43843026157983
// MI455X (gfx1250) — compile-verified
//
#include <hip/hip_runtime.h>
#include <hip/hip_bf16.h>

// MPNN forward for MI455X (gfx1250, wave32).
// All GEMMs on the matrix pipe via V_WMMA_F32_16X16X4_F32 (f32 to match the
// reference numerically). Each wave computes a 32-row x 96-col tile (M=2
// sub-tiles of 16). Weight B-fragments are pre-packed in workspace in lane
// order so the inner loop does ONE coalesced global_load_b64 per fragment;
// A and B fragments are double-buffered in registers across K-steps so WMMAs
// of step k overlap the loads of step k+1 (no full loadcnt drain per WMMA).
// ReLU intermediate goes through a padded static LDS tile (C->A layout).
// Workspace (floats): h[N*96] | agg[N*96] | deg[N] | gsum[G*96] | gcnt[G] |
// packed weights (~264K floats)  => ~39.7 MB total.

typedef float v2f __attribute__((ext_vector_type(2)));
typedef float v8f __attribute__((ext_vector_type(8)));

namespace {
constexpr int Nn   = 50000;
constexpr int Ee   = 800000;
constexpr int NF_  = 64;
constexpr int EF_  = 32;
constexpr int H_   = 96;
constexpr int OUT_ = 64;
constexpr int L_   = 4;
constexpr int G_   = 64;
constexpr int NT   = H_ / 16;    // 6 column tiles
constexpr int NTO  = OUT_ / 16;  // 4 column tiles (head stage 2)
constexpr int LDSW = 100;        // padded LDS row stride (floats)
}

__device__ __forceinline__ v8f wmma4(v2f a, v2f b, v8f c) {
  return __builtin_amdgcn_wmma_f32_16x16x4_f32(false, a, false, b,
                                               (short)0, c, false, false);
}

__device__ __forceinline__ v8f splat8(float x) {
  v8f r;
#pragma unroll
  for (int i = 0; i < 8; ++i) r[i] = x;
  return r;
}

// Dual-M-tile GEMM: acc[t][nt] += A_t x B, K = 4*KT, double-buffered frags.
template <int KT, int NTn, class LA, class LB>
__device__ __forceinline__ void gemm_dual(LA ldA, LB ldB,
                                          v8f (&acc0)[NTn], v8f (&acc1)[NTn]) {
  v2f a0 = ldA(0, 0), a1 = ldA(0, 1);
  v2f b[NTn];
#pragma unroll
  for (int nt = 0; nt < NTn; ++nt) b[nt] = ldB(0, nt);
#pragma unroll
  for (int kk = 0; kk < KT; ++kk) {
    v2f na0, na1, nb[NTn];
    if (kk + 1 < KT) {
      na0 = ldA(kk + 1, 0); na1 = ldA(kk + 1, 1);
#pragma unroll
      for (int nt = 0; nt < NTn; ++nt) nb[nt] = ldB(kk + 1, nt);
    } else {
      na0 = a0; na1 = a1;
#pragma unroll
      for (int nt = 0; nt < NTn; ++nt) nb[nt] = b[nt];
    }
#pragma unroll
    for (int nt = 0; nt < NTn; ++nt) {
      acc0[nt] = wmma4(a0, b[nt], acc0[nt]);
      acc1[nt] = wmma4(a1, b[nt], acc1[nt]);
    }
    a0 = na0; a1 = na1;
#pragma unroll
    for (int nt = 0; nt < NTn; ++nt) b[nt] = nb[nt];
  }
}

// Single-M-tile GEMM with the same double-buffer structure.
template <int KT, int NTn, class LA, class LB>
__device__ __forceinline__ void gemm_one(LA ldA, LB ldB, v8f (&acc)[NTn]) {
  v2f a = ldA(0);
  v2f b[NTn];
#pragma unroll
  for (int nt = 0; nt < NTn; ++nt) b[nt] = ldB(0, nt);
#pragma unroll
  for (int kk = 0; kk < KT; ++kk) {
    v2f na, nb[NTn];
    if (kk + 1 < KT) {
      na = ldA(kk + 1);
#pragma unroll
      for (int nt = 0; nt < NTn; ++nt) nb[nt] = ldB(kk + 1, nt);
    } else {
      na = a;
#pragma unroll
      for (int nt = 0; nt < NTn; ++nt) nb[nt] = b[nt];
    }
#pragma unroll
    for (int nt = 0; nt < NTn; ++nt) acc[nt] = wmma4(a, b[nt], acc[nt]);
    a = na;
#pragma unroll
    for (int nt = 0; nt < NTn; ++nt) b[nt] = nb[nt];
  }
}

// ------------------------------------------------------------- pack B ------
// P[((kk*NTn + nt)*32 + lane)] = {W[k0][n], W[k0+1][n]},
// k0 = kk*4 + (lane>>4)*2, n = nt*16 + (lane&15). One b64 per fragment later.
__global__ void pack_kernel(const float* __restrict__ W, float* __restrict__ P,
                            int KT, int NTn, int Nw) {
  const int idx = blockIdx.x * 256 + threadIdx.x;
  const int total = KT * NTn * 32;
  if (idx >= total) return;
  const int lane = idx & 31;
  const int fn = idx >> 5;
  const int nt = fn % NTn;
  const int kk = fn / NTn;
  const int k0 = kk * 4 + (lane >> 4) * 2;
  const int n = nt * 16 + (lane & 15);
  v2f b; b.x = W[k0 * Nw + n]; b.y = W[(k0 + 1) * Nw + n];
  ((v2f*)P)[idx] = b;
}

// ---------------------------------------------------------------- embed ----
__global__ __launch_bounds__(128) void embed_kernel(
    const float* __restrict__ x,
    const float* __restrict__ P1, const float* __restrict__ B1,
    const float* __restrict__ P2, const float* __restrict__ B2,
    float* __restrict__ h) {
  __shared__ float lds[4][16][LDSW];
  const int lane = threadIdx.x & 31, wave = threadIdx.x >> 5;
  const int grp = blockIdx.x * 4 + wave;
  const int ngrp = (Nn + 31) >> 5;
  if (grp >= ngrp) return;                       // wave-uniform
  const int ln = lane & 15, kh = (lane >> 4) * 2, rh = (lane >> 4) * 8;
  const int base = grp * 32;
  const int i0 = base + ln, i1 = base + 16 + ln;
  const float* px0 = x + (size_t)(i0 < Nn ? i0 : 0) * NF_;
  const float* px1 = x + (size_t)(i1 < Nn ? i1 : 0) * NF_;
  const v2f* pW1 = (const v2f*)P1;
  const v2f* pW2 = (const v2f*)P2;

  v8f acc0[NT], acc1[NT];
#pragma unroll
  for (int nt = 0; nt < NT; ++nt) {
    const float bb = B1[nt * 16 + ln];
    acc0[nt] = splat8(bb); acc1[nt] = splat8(bb);
  }
  gemm_dual<NF_ / 4, NT>(
      [&](int kk, int t) -> v2f {
        const float* p = t ? px1 : px0;
        return *(const v2f*)(p + kk * 4 + kh);
      },
      [&](int kk, int nt) -> v2f { return pW1[(kk * NT + nt) * 32 + lane]; },
      acc0, acc1);

#pragma unroll
  for (int t = 0; t < 2; ++t) {
#pragma unroll
    for (int nt = 0; nt < NT; ++nt)
#pragma unroll
      for (int v = 0; v < 8; ++v) {
        const float q = t ? acc1[nt][v] : acc0[nt][v];
        lds[wave][v + rh][nt * 16 + ln] = q > 0.f ? q : 0.f;
      }
    v8f acc2[NT];
#pragma unroll
    for (int nt = 0; nt < NT; ++nt) acc2[nt] = splat8(B2[nt * 16 + ln]);
    gemm_one<H_ / 4, NT>(
        [&](int kk) -> v2f { return *(const v2f*)&lds[wave][ln][kk * 4 + kh]; },
        [&](int kk, int nt) -> v2f { return pW2[(kk * NT + nt) * 32 + lane]; },
        acc2);
#pragma unroll
    for (int v = 0; v < 8; ++v) {
      const int i2 = base + t * 16 + v + rh;
      if (i2 < Nn) {
#pragma unroll
        for (int nt = 0; nt < NT; ++nt)
          h[(size_t)i2 * H_ + nt * 16 + ln] = acc2[nt][v];
      }
    }
  }
}

// -------------------------------------------------------------- message ----
__global__ __launch_bounds__(128) void msg_kernel(
    const float* __restrict__ h, const float* __restrict__ ea,
    const int* __restrict__ src, const int* __restrict__ tgt,
    const float* __restrict__ P1, const float* __restrict__ B1,
    const float* __restrict__ P2, const float* __restrict__ B2,
    float* __restrict__ agg) {
  __shared__ float lds[4][16][LDSW];
  const int lane = threadIdx.x & 31, wave = threadIdx.x >> 5;
  const int grp = blockIdx.x * 4 + wave;
  const int ngrp = (Ee + 31) >> 5;
  if (grp >= ngrp) return;
  const int ln = lane & 15, kh = (lane >> 4) * 2, rh = (lane >> 4) * 8;
  const int base = grp * 32;
  const int e0 = base + ln, e1 = base + 16 + ln;
  const int ec0 = e0 < Ee ? e0 : 0, ec1 = e1 < Ee ? e1 : 0;
  const float* pt0 = h + (size_t)tgt[ec0] * H_;
  const float* ps0 = h + (size_t)src[ec0] * H_;
  const float* pe0 = ea + (size_t)ec0 * EF_;
  const float* pt1 = h + (size_t)tgt[ec1] * H_;
  const float* ps1 = h + (size_t)src[ec1] * H_;
  const float* pe1 = ea + (size_t)ec1 * EF_;
  const v2f* pW1 = (const v2f*)P1;
  const v2f* pW2 = (const v2f*)P2;

  v8f acc0[NT], acc1[NT];
#pragma unroll
  for (int nt = 0; nt < NT; ++nt) {
    const float bb = B1[nt * 16 + ln];
    acc0[nt] = splat8(bb); acc1[nt] = splat8(bb);
  }
  gemm_dual<(2 * H_ + EF_) / 4, NT>(            // 56 K-steps
      [&](int kk, int t) -> v2f {
        const int k0 = kk * 4 + kh;
        if (kk < 24)      return *(const v2f*)((t ? pt1 : pt0) + k0);
        else if (kk < 48) return *(const v2f*)((t ? ps1 : ps0) + (k0 - 96));
        else              return *(const v2f*)((t ? pe1 : pe0) + (k0 - 192));
      },
      [&](int kk, int nt) -> v2f { return pW1[(kk * NT + nt) * 32 + lane]; },
      acc0, acc1);

#pragma unroll
  for (int t = 0; t < 2; ++t) {
#pragma unroll
    for (int nt = 0; nt < NT; ++nt)
#pragma unroll
      for (int v = 0; v < 8; ++v) {
        const float q = t ? acc1[nt][v] : acc0[nt][v];
        lds[wave][v + rh][nt * 16 + ln] = q > 0.f ? q : 0.f;
      }
    v8f acc2[NT];
#pragma unroll
    for (int nt = 0; nt < NT; ++nt) acc2[nt] = splat8(B2[nt * 16 + ln]);
    gemm_one<H_ / 4, NT>(
        [&](int kk) -> v2f { return *(const v2f*)&lds[wave][ln][kk * 4 + kh]; },
        [&](int kk, int nt) -> v2f { return pW2[(kk * NT + nt) * 32 + lane]; },
        acc2);
    // scatter-add messages into agg[tgt]
#pragma unroll
    for (int v = 0; v < 8; ++v) {
      const int er = base + t * 16 + v + rh;
      if (er < Ee) {
        const int tg = tgt[er];
#pragma unroll
        for (int nt = 0; nt < NT; ++nt)
          atomicAdd(agg + (size_t)tg * H_ + nt * 16 + ln, acc2[nt][v]);
      }
    }
  }
}

// --------------------------------------------------------------- update ----
__global__ __launch_bounds__(128) void upd_kernel(
    float* __restrict__ h, const float* __restrict__ agg,
    const float* __restrict__ deg,
    const float* __restrict__ P1, const float* __restrict__ B1,
    const float* __restrict__ P2, const float* __restrict__ B2) {
  __shared__ float lds[4][16][LDSW];
  const int lane = threadIdx.x & 31, wave = threadIdx.x >> 5;
  const int grp = blockIdx.x * 4 + wave;
  const int ngrp = (Nn + 31) >> 5;
  if (grp >= ngrp) return;
  const int ln = lane & 15, kh = (lane >> 4) * 2, rh = (lane >> 4) * 8;
  const int base = grp * 32;
  const int i0 = base + ln, i1 = base + 16 + ln;
  const int ic0 = i0 < Nn ? i0 : 0, ic1 = i1 < Nn ? i1 : 0;
  const float* ph0 = h + (size_t)ic0 * H_;
  const float* pa0 = agg + (size_t)ic0 * H_;
  const float* ph1 = h + (size_t)ic1 * H_;
  const float* pa1 = agg + (size_t)ic1 * H_;
  const float rd0 = 1.0f / fmaxf(deg[ic0], 1.0f);
  const float rd1 = 1.0f / fmaxf(deg[ic1], 1.0f);
  const v2f* pW1 = (const v2f*)P1;
  const v2f* pW2 = (const v2f*)P2;

  v8f acc0[NT], acc1[NT];
#pragma unroll
  for (int nt = 0; nt < NT; ++nt) {
    const float bb = B1[nt * 16 + ln];
    acc0[nt] = splat8(bb); acc1[nt] = splat8(bb);
  }
  gemm_dual<(2 * H_) / 4, NT>(                   // 48 K-steps
      [&](int kk, int t) -> v2f {
        const int k0 = kk * 4 + kh;
        if (kk < 24) return *(const v2f*)((t ? ph1 : ph0) + k0);
        v2f a = *(const v2f*)((t ? pa1 : pa0) + (k0 - 96));
        const float rd = t ? rd1 : rd0;
        a.x *= rd; a.y *= rd;
        return a;
      },
      [&](int kk, int nt) -> v2f { return pW1[(kk * NT + nt) * 32 + lane]; },
      acc0, acc1);

#pragma unroll
  for (int t = 0; t < 2; ++t) {
#pragma unroll
    for (int nt = 0; nt < NT; ++nt)
#pragma unroll
      for (int v = 0; v < 8; ++v) {
        const float q = t ? acc1[nt][v] : acc0[nt][v];
        lds[wave][v + rh][nt * 16 + ln] = q > 0.f ? q : 0.f;
      }
    v8f acc2[NT];
#pragma unroll
    for (int nt = 0; nt < NT; ++nt) acc2[nt] = splat8(B2[nt * 16 + ln]);
    gemm_one<H_ / 4, NT>(
        [&](int kk) -> v2f { return *(const v2f*)&lds[wave][ln][kk * 4 + kh]; },
        [&](int kk, int nt) -> v2f { return pW2[(kk * NT + nt) * 32 + lane]; },
        acc2);
#pragma unroll
    for (int v = 0; v < 8; ++v) {
      const int i2 = base + t * 16 + v + rh;
      if (i2 < Nn) {
#pragma unroll
        for (int nt = 0; nt < NT; ++nt)
          h[(size_t)i2 * H_ + nt * 16 + ln] = acc2[nt][v];
      }
    }
  }
}

// ----------------------------------------------------------------- head ----
__global__ __launch_bounds__(128) void head_kernel(
    const float* __restrict__ gsum, const float* __restrict__ gcnt,
    const float* __restrict__ P1, const float* __restrict__ B1,
    const float* __restrict__ P2, const float* __restrict__ B2,
    float* __restrict__ out) {
  __shared__ float lds[4][16][LDSW];
  const int lane = threadIdx.x & 31, wave = threadIdx.x >> 5;
  const int tile = wave;                         // G=64 -> exactly 4 tiles
  const int ln = lane & 15, kh = (lane >> 4) * 2, rh = (lane >> 4) * 8;
  const int g = tile * 16 + ln;
  const float* pg = gsum + (size_t)g * H_;
  const float sc = 1.0f / fmaxf(gcnt[g], 1.0f);
  const v2f* pW1 = (const v2f*)P1;
  const v2f* pW2 = (const v2f*)P2;

  v8f acc[NT];
#pragma unroll
  for (int nt = 0; nt < NT; ++nt) acc[nt] = splat8(B1[nt * 16 + ln]);
  gemm_one<H_ / 4, NT>(
      [&](int kk) -> v2f {
        v2f a = *(const v2f*)(pg + kk * 4 + kh);
        a.x *= sc; a.y *= sc;
        return a;
      },
      [&](int kk, int nt) -> v2f { return pW1[(kk * NT + nt) * 32 + lane]; },
      acc);
#pragma unroll
  for (int nt = 0; nt < NT; ++nt)
#pragma unroll
    for (int v = 0; v < 8; ++v) {
      const float q = acc[nt][v];
      lds[wave][v + rh][nt * 16 + ln] = q > 0.f ? q : 0.f;
    }
  v8f acc2[NTO];
#pragma unroll
  for (int nt = 0; nt < NTO; ++nt) acc2[nt] = splat8(B2[nt * 16 + ln]);
  gemm_one<H_ / 4, NTO>(
      [&](int kk) -> v2f { return *(const v2f*)&lds[wave][ln][kk * 4 + kh]; },
      [&](int kk, int nt) -> v2f { return pW2[(kk * NTO + nt) * 32 + lane]; },
      acc2);
#pragma unroll
  for (int v = 0; v < 8; ++v) {
    const int g2 = tile * 16 + v + rh;
#pragma unroll
    for (int nt = 0; nt < NTO; ++nt)
      out[(size_t)g2 * OUT_ + nt * 16 + ln] = acc2[nt][v];
  }
}

// -------------------------------------------------------------- helpers ----
__global__ void deg_kernel(const int* __restrict__ tgt, float* __restrict__ deg) {
  const int e = blockIdx.x * 256 + threadIdx.x;
  if (e < Ee) atomicAdd(&deg[tgt[e]], 1.0f);
}

__global__ void pool_kernel(const float* __restrict__ h,
                            const int* __restrict__ batch,
                            float* __restrict__ gsum, float* __restrict__ gcnt) {
  const int i = blockIdx.x * 256 + threadIdx.x;
  if (i >= Nn) return;
  const int b = batch[i];
  const float* ph = h + (size_t)i * H_;
#pragma unroll 4
  for (int f = 0; f < H_; ++f) atomicAdd(&gsum[(size_t)b * H_ + f], ph[f]);
  atomicAdd(&gcnt[b], 1.0f);
}

// --------------------------------------------------------------- launch ----
static inline void launch_pack(const float* W, float* P, int KT, int NTn,
                               int Nw, hipStream_t stream) {
  const int total = KT * NTn * 32;
  pack_kernel<<<(total + 255) / 256, 256, 0, stream>>>(W, P, KT, NTn, Nw);
}

extern "C" void kernel_launch(void* const* d_in, const int* in_sizes, int n_in,
                              void* d_out, int out_size, void* d_ws, size_t ws_size,
                              hipStream_t stream) {
  const float* x       = (const float*)d_in[0];
  const float* ea      = (const float*)d_in[1];
  const int*   eidx    = (const int*)d_in[2];
  const int*   src     = eidx;            // edge_index[0]
  const int*   tgt     = eidx + Ee;       // edge_index[1]
  const int*   batch   = (const int*)d_in[3];
  const float* emb_w1  = (const float*)d_in[4];
  const float* emb_b1  = (const float*)d_in[5];
  const float* emb_w2  = (const float*)d_in[6];
  const float* emb_b2  = (const float*)d_in[7];
  const float* msg_w1  = (const float*)d_in[8];
  const float* msg_b1  = (const float*)d_in[9];
  const float* msg_w2  = (const float*)d_in[10];
  const float* msg_b2  = (const float*)d_in[11];
  const float* upd_w1  = (const float*)d_in[12];
  const float* upd_b1  = (const float*)d_in[13];
  const float* upd_w2  = (const float*)d_in[14];
  const float* upd_b2  = (const float*)d_in[15];
  const float* head_w1 = (const float*)d_in[16];
  const float* head_b1 = (const float*)d_in[17];
  const float* head_w2 = (const float*)d_in[18];
  const float* head_b2 = (const float*)d_in[19];

  // Fragment-step counts
  constexpr int KT_E1 = NF_ / 4;             // 16
  constexpr int KT_M1 = (2 * H_ + EF_) / 4;  // 56
  constexpr int KT_U1 = (2 * H_) / 4;        // 48
  constexpr int KT_H  = H_ / 4;              // 24

  float* ws   = (float*)d_ws;
  float* h    = ws;
  float* agg  = h + (size_t)Nn * H_;
  float* deg  = agg + (size_t)Nn * H_;
  float* gsum = deg + Nn;
  float* gcnt = gsum + (size_t)G_ * H_;
  float* pk   = gcnt + G_;                   // packed weights region
  float* p_e1 = pk;                  float* p_e2 = p_e1 + KT_E1 * NT * 64;
  float* p_m1 = p_e2 + KT_H * NT * 64;                 // [L][56*6*64]
  float* p_m2 = p_m1 + (size_t)L_ * KT_M1 * NT * 64;   // [L][24*6*64]
  float* p_u1 = p_m2 + (size_t)L_ * KT_H * NT * 64;    // [L][48*6*64]
  float* p_u2 = p_u1 + (size_t)L_ * KT_U1 * NT * 64;   // [L][24*6*64]
  float* p_h1 = p_u2 + (size_t)L_ * KT_H * NT * 64;
  float* p_h2 = p_h1 + KT_H * NT * 64;

  // Pack all weight matrices into fragment order (tiny, L2-resident).
  launch_pack(emb_w1, p_e1, KT_E1, NT, H_, stream);
  launch_pack(emb_w2, p_e2, KT_H, NT, H_, stream);
  for (int l = 0; l < L_; ++l) {
    launch_pack(msg_w1 + (size_t)l * (2 * H_ + EF_) * H_,
                p_m1 + (size_t)l * KT_M1 * NT * 64, KT_M1, NT, H_, stream);
    launch_pack(msg_w2 + (size_t)l * H_ * H_,
                p_m2 + (size_t)l * KT_H * NT * 64, KT_H, NT, H_, stream);
    launch_pack(upd_w1 + (size_t)l * (2 * H_) * H_,
                p_u1 + (size_t)l * KT_U1 * NT * 64, KT_U1, NT, H_, stream);
    launch_pack(upd_w2 + (size_t)l * H_ * H_,
                p_u2 + (size_t)l * KT_H * NT * 64, KT_H, NT, H_, stream);
  }
  launch_pack(head_w1, p_h1, KT_H, NT, H_, stream);
  launch_pack(head_w2, p_h2, KT_H, NTO, OUT_, stream);

  // zero deg | gsum | gcnt (contiguous)
  hipMemsetAsync(deg, 0, (size_t)(Nn + G_ * H_ + G_) * sizeof(float), stream);

  const int nodeBlocks = (((Nn + 31) / 32) + 3) / 4;
  const int edgeBlocks = (((Ee + 31) / 32) + 3) / 4;   // = 6250 exact

  embed_kernel<<<nodeBlocks, 128, 0, stream>>>(x, p_e1, emb_b1, p_e2, emb_b2, h);
  deg_kernel<<<(Ee + 255) / 256, 256, 0, stream>>>(tgt, deg);

  for (int l = 0; l < L_; ++l) {
    hipMemsetAsync(agg, 0, (size_t)Nn * H_ * sizeof(float), stream);
    msg_kernel<<<edgeBlocks, 128, 0, stream>>>(
        h, ea, src, tgt,
        p_m1 + (size_t)l * KT_M1 * NT * 64, msg_b1 + (size_t)l * H_,
        p_m2 + (size_t)l * KT_H * NT * 64,  msg_b2 + (size_t)l * H_,
        agg);
    upd_kernel<<<nodeBlocks, 128, 0, stream>>>(
        h, agg, deg,
        p_u1 + (size_t)l * KT_U1 * NT * 64, upd_b1 + (size_t)l * H_,
        p_u2 + (size_t)l * KT_H * NT * 64,  upd_b2 + (size_t)l * H_);
  }

  pool_kernel<<<(Nn + 255) / 256, 256, 0, stream>>>(h, batch, gsum, gcnt);
  head_kernel<<<1, 128, 0, stream>>>(gsum, gcnt, p_h1, head_b1,
                                     p_h2, head_b2, (float*)d_out);
}